// _SDNLayer_15513421873205
// MI455X (gfx1250) — compile-verified
//
#include <hip/hip_runtime.h>

typedef __attribute__((ext_vector_type(2))) float v2f;
typedef __attribute__((ext_vector_type(8))) float v8f;

#define B_  8
#define C_  32
#define H_  512
#define W_  512
#define TC  96      // 3*C

// LDS row strides (padded for bank-conflict avoidance on 64 banks)
#define PS 34       // prev/cur tiles: stride 34 -> lanes m=0..15 hit distinct banks, 8B aligned
#define SS 104      // Si/Sh tiles:   8*104 mod 64 = 0? 832 mod 64 = 0 -> no; 104*8=832, 832%64=0... use offset via distinct cols anyway

// Column-0 passthrough: out[:,:,:,0] = states[:,:,:,0]
__global__ void sdn_copy_col0(const float* __restrict__ states, float* __restrict__ out) {
    int i = blockIdx.x * blockDim.x + threadIdx.x;   // flat over B*C*H
    if (i < B_ * C_ * H_)
        out[(size_t)i * W_] = states[(size_t)i * W_];
}

__global__ __launch_bounds__(192) void sdn_col_kernel(
    const float* __restrict__ states,   // (B,C,H,W)
    const float* __restrict__ w_ih,     // (96,96) row-major (out, in)
    const float* __restrict__ b_ih,     // (96)
    const float* __restrict__ w_hh,     // (96,32)
    const float* __restrict__ b_hh,     // (96)
    float* __restrict__ out,            // (B,C,H,W)
    int w)
{
    __shared__ float prevX[18 * PS];    // prev-output rows h0-1 .. h0+16, 32 ch each
    __shared__ float curX [16 * PS];    // current input rows h0 .. h0+15
    __shared__ float Si   [16 * SS];    // i_vals pre-activation (16 rows x 96)
    __shared__ float Sh   [16 * SS];    // h_vals pre-activation

    const int tid = threadIdx.x;
    const int blk = blockIdx.x;         // 256 blocks: 32 h-tiles per batch
    const int b   = blk >> 5;
    const int h0  = (blk & 31) << 4;

    // ---- Phase 0: cooperative loads (strided global -> LDS) ----
    for (int i = tid; i < 18 * 32; i += 192) {
        int r = i >> 5, c = i & 31;
        int h = h0 - 1 + r;             // r=0 -> h0-1 (zero pad at H edges)
        float v = 0.0f;
        if (h >= 0 && h < H_)
            v = out[(((size_t)b * C_ + c) * H_ + h) * W_ + (w - 1)];
        prevX[r * PS + c] = v;
    }
    for (int i = tid; i < 16 * 32; i += 192) {
        int r = i >> 5, c = i & 31;
        curX[r * PS + c] = states[(((size_t)b * C_ + c) * H_ + (h0 + r)) * W_ + w];
    }
    __syncthreads();

    // ---- Phase 1: WMMA GEMMs (uniform EXEC across all 6 waves) ----
    {
        const int wv    = tid >> 5;     // wave id 0..5 -> N-tile
        const int lane  = tid & 31;
        const int half  = lane >> 4;    // lane half selects K pair / M+8
        const int lm    = lane & 15;    // A: M row; B: N col
        const int ntile = wv << 4;
        const int n     = ntile + lm;

        // GEMM1: i_vals tile = X(16x96) @ W_ih^T, X[m][k] = prevX[m + k/32][k%32]
        v8f acc = {};
        #pragma unroll
        for (int kk = 0; kk < 24; ++kk) {
            const int k0 = kk * 4 + half * 2;            // even -> (k0,k0+1) share 32-block
            v2f a;
            a.x = prevX[(lm + (k0 >> 5)) * PS + (k0 & 31)];
            a.y = prevX[(lm + (k0 >> 5)) * PS + ((k0 & 31) + 1)];
            v2f bb = *(const v2f*)(w_ih + (size_t)n * TC + k0);   // B[k][n] = w_ih[n][k]
            acc = __builtin_amdgcn_wmma_f32_16x16x4_f32(
                    /*neg_a=*/false, a, /*neg_b=*/false, bb,
                    /*c_mod=*/(short)0, acc, /*reuse_a=*/false, /*reuse_b=*/false);
        }
        const float bias1 = b_ih[n];
        #pragma unroll
        for (int j = 0; j < 8; ++j)                       // C/D: M = j + 8*half, N = lm
            Si[(j + half * 8) * SS + n] = acc[j] + bias1;

        // GEMM2: h_vals tile = cur(16x32) @ W_hh^T
        v8f acc2 = {};
        #pragma unroll
        for (int kk = 0; kk < 8; ++kk) {
            const int k0 = kk * 4 + half * 2;
            v2f a;
            a.x = curX[lm * PS + k0];
            a.y = curX[lm * PS + k0 + 1];
            v2f bb = *(const v2f*)(w_hh + (size_t)n * C_ + k0);
            acc2 = __builtin_amdgcn_wmma_f32_16x16x4_f32(
                    false, a, false, bb, (short)0, acc2, false, false);
        }
        const float bias2 = b_hh[n];
        #pragma unroll
        for (int j = 0; j < 8; ++j)
            Sh[(j + half * 8) * SS + n] = acc2[j] + bias2;
    }
    __syncthreads();

    // ---- Phase 2: GRU gates + output column write ----
    for (int i = tid; i < 16 * 32; i += 192) {
        int r = i >> 5, c = i & 31;
        float ri = Si[r * SS + c],      rh = Sh[r * SS + c];
        float zi = Si[r * SS + 32 + c], zh = Sh[r * SS + 32 + c];
        float ni = Si[r * SS + 64 + c], nh = Sh[r * SS + 64 + c];
        float rg = 1.0f / (1.0f + __expf(-(ri + rh)));
        float zg = 1.0f / (1.0f + __expf(-(zi + zh)));
        float ng = tanhf(ni + rg * nh);
        float cv = curX[r * PS + c];
        out[(((size_t)b * C_ + c) * H_ + (h0 + r)) * W_ + w] = ng * zg + cv * (1.0f - zg);
    }
}

extern "C" void kernel_launch(void* const* d_in, const int* in_sizes, int n_in,
                              void* d_out, int out_size, void* d_ws, size_t ws_size,
                              hipStream_t stream) {
    const float* states = (const float*)d_in[0];
    const float* w_ih   = (const float*)d_in[1];
    const float* b_ih   = (const float*)d_in[2];
    const float* w_hh   = (const float*)d_in[3];
    const float* b_hh   = (const float*)d_in[4];
    float* out = (float*)d_out;

    // Column 0 is a passthrough of the input.
    sdn_copy_col0<<<(B_ * C_ * H_ + 255) / 256, 256, 0, stream>>>(states, out);

    // Sequential scan over W: stream ordering provides the w-1 -> w dependency.
    for (int w = 1; w < W_; ++w)
        sdn_col_kernel<<<256, 192, 0, stream>>>(states, w_ih, b_ih, w_hh, b_hh, out, w);
}